// QuantizationLayerVoxGrid_65395172049713
// MI455X (gfx1250) — compile-verified
//
#include <hip/hip_runtime.h>
#include <stdint.h>

// Problem constants (match reference)
#define B_ 16
#define C_ 9
#define H_ 260
#define W_ 346

#define NBIN 16384        // value-bins per batch for bucket sort
#define TILE 1024         // events per block tile (1024*20B = 20480B LDS)
#define HTHREADS 256
#define CAP 256           // max segment length handled in LDS (mean ~30, Poisson)

// ---------- CDNA5 async global->LDS helpers (gfx1250) ----------
// Low 32 bits of a flat pointer into the LDS aperture are the wave-relative
// LDS byte offset.
__device__ __forceinline__ uint32_t lds_off32(const void* p) {
  return (uint32_t)(uintptr_t)p;
}
__device__ __forceinline__ void async_copy16(uint32_t lds_off, const void* gsrc) {
  // GLOBAL_LOAD_ASYNC_TO_LDS_B128: per-lane 16B global -> LDS, tracked by ASYNCcnt
  asm volatile("global_load_async_to_lds_b128 %0, %1, off"
               :: "v"(lds_off), "v"((uint64_t)(uintptr_t)gsrc)
               : "memory");
}
__device__ __forceinline__ void wait_async0() {
  asm volatile("s_wait_asynccnt 0" ::: "memory");
}

// ---------- Pass 1: per-batch histogram of t into NBIN uniform bins ----------
__global__ void __launch_bounds__(HTHREADS)
k_hist(const float* __restrict__ ev, uint32_t N, uint32_t* __restrict__ hist) {
  __shared__ __align__(16) float sm[TILE * 5];
  const uint32_t tile0 = (uint32_t)blockIdx.x * TILE;
  if (tile0 >= N) return;
  const uint32_t nev = (N - tile0 < TILE) ? (N - tile0) : TILE;
  const uint32_t nbytes = nev * 20u;
  const char* src = (const char*)ev + (size_t)tile0 * 20u;
  const uint32_t tid = threadIdx.x;
  const uint32_t lds0 = lds_off32(sm);

  const uint32_t n16 = nbytes >> 4;
  for (uint32_t c = tid; c < n16; c += HTHREADS)
    async_copy16(lds0 + c * 16u, src + (size_t)c * 16u);
  // ragged tail (only if nev % 4 != 0) via plain dword copies
  for (uint32_t o = (n16 << 4) + tid * 4u; o < nbytes; o += HTHREADS * 4u)
    *(uint32_t*)((char*)sm + o) = *(const uint32_t*)(src + o);
  wait_async0();
  __syncthreads();

  for (uint32_t e = tid; e < nev; e += HTHREADS) {
    float t = sm[e * 5 + 2];
    int   b = (int)sm[e * 5 + 4];
    int bin = (int)(t * (float)NBIN);
    bin = bin < 0 ? 0 : (bin > NBIN - 1 ? NBIN - 1 : bin);
    atomicAdd(&hist[(size_t)b * NBIN + bin], 1u);
  }
}

// ---------- Pass 2: per-batch exclusive scan of 16384 bins ----------
__global__ void __launch_bounds__(1024)
k_scan(const uint32_t* __restrict__ hist, uint32_t* __restrict__ binStart,
       uint32_t* __restrict__ counts, uint32_t* __restrict__ unitArr) {
  __shared__ uint32_t sdata[1024];
  const int b = blockIdx.x;
  const int tid = threadIdx.x;
  const uint32_t* h = hist + (size_t)b * NBIN;
  uint32_t* s = binStart + (size_t)b * NBIN;
  constexpr int PER = NBIN / 1024; // 16
  uint32_t v[PER];
  uint32_t sum = 0;
  const int base = tid * PER;
#pragma unroll
  for (int i = 0; i < PER; ++i) { v[i] = h[base + i]; sum += v[i]; }
  sdata[tid] = sum;
  __syncthreads();
  for (int off = 1; off < 1024; off <<= 1) {
    uint32_t t = (tid >= off) ? sdata[tid - off] : 0u;
    __syncthreads();
    sdata[tid] += t;
    __syncthreads();
  }
  uint32_t run = sdata[tid] - sum; // exclusive prefix of this thread's chunk
#pragma unroll
  for (int i = 0; i < PER; ++i) { s[base + i] = run; run += v[i]; }
  if (tid == 1023) {
    uint32_t total = sdata[1023];
    counts[b] = total;
    uint32_t u = total / C_;
    unitArr[b] = u ? u : 1u;      // reference: max(counts//C, 1)
  }
}

// ---------- Pass 3: cross-batch offsets (16 values) ----------
__global__ void k_batchoff(const uint32_t* __restrict__ counts,
                           uint32_t* __restrict__ batchOff) {
  if (threadIdx.x == 0 && blockIdx.x == 0) {
    uint32_t acc = 0;
    for (int i = 0; i < B_; ++i) { batchOff[i] = acc; acc += counts[i]; }
  }
}

// ---------- Pass 4: scatter {index, key} into bin segments ----------
__global__ void __launch_bounds__(HTHREADS)
k_scatter(const float* __restrict__ ev, uint32_t N,
          const uint32_t* __restrict__ binStart, const uint32_t* __restrict__ batchOff,
          uint32_t* __restrict__ cursor, uint2* __restrict__ pairs) {
  __shared__ __align__(16) float sm[TILE * 5];
  const uint32_t tile0 = (uint32_t)blockIdx.x * TILE;
  if (tile0 >= N) return;
  const uint32_t nev = (N - tile0 < TILE) ? (N - tile0) : TILE;
  const uint32_t nbytes = nev * 20u;
  const char* src = (const char*)ev + (size_t)tile0 * 20u;
  const uint32_t tid = threadIdx.x;
  const uint32_t lds0 = lds_off32(sm);

  const uint32_t n16 = nbytes >> 4;
  for (uint32_t c = tid; c < n16; c += HTHREADS)
    async_copy16(lds0 + c * 16u, src + (size_t)c * 16u);
  for (uint32_t o = (n16 << 4) + tid * 4u; o < nbytes; o += HTHREADS * 4u)
    *(uint32_t*)((char*)sm + o) = *(const uint32_t*)(src + o);
  wait_async0();
  __syncthreads();

  for (uint32_t e = tid; e < nev; e += HTHREADS) {
    float t = sm[e * 5 + 2];
    int   b = (int)sm[e * 5 + 4];
    int bin = (int)(t * (float)NBIN);
    bin = bin < 0 ? 0 : (bin > NBIN - 1 ? NBIN - 1 : bin);
    uint32_t key = __float_as_uint(t);        // positive floats: bit order == value order
    uint32_t pos = atomicAdd(&cursor[(size_t)b * NBIN + bin], 1u);
    uint32_t slot = batchOff[b] + binStart[(size_t)b * NBIN + bin] + pos;
    pairs[slot] = make_uint2(tile0 + e, key); // .x = global event index, .y = key
  }
}

// ---------- Pass 5: per-bin exact rank -> ibin -> scatter-add output ----------
__device__ __forceinline__ void emit_event(const float* __restrict__ ev,
                                           float* __restrict__ out,
                                           uint32_t idx, int b,
                                           uint32_t rank, uint32_t unit) {
  uint32_t ib = rank / unit;
  if (ib < C_) {
    const float* e = ev + (size_t)idx * 5;
    int x = (int)e[0];
    int y = (int)e[1];
    int p = (int)e[3];
    // output layout (C, B, 2, H, W)
    size_t o = ((((size_t)ib * B_ + (size_t)b) * 2 + (size_t)p) * H_ + (size_t)y) * W_ + (size_t)x;
    atomicAdd(out + o, 1.0f);
  }
}

__global__ void __launch_bounds__(256)
k_binsort(const float* __restrict__ ev, const uint2* __restrict__ pairs,
          const uint32_t* __restrict__ binStart, const uint32_t* __restrict__ counts,
          const uint32_t* __restrict__ unitArr, const uint32_t* __restrict__ batchOff,
          float* __restrict__ out) {
  __shared__ uint2 seg[8][CAP];                 // 16 KB
  const int wave = threadIdx.x >> 5;            // wave32
  const int lane = threadIdx.x & 31;
  const int gwave = blockIdx.x * 8 + wave;      // one wave per (batch, bin)
  const int b   = gwave >> 14;                  // / NBIN
  const int bin = gwave & (NBIN - 1);

  const uint32_t segS = binStart[(size_t)b * NBIN + bin];
  const uint32_t segE = (bin == NBIN - 1) ? counts[b] : binStart[(size_t)b * NBIN + bin + 1];
  const uint32_t len = segE - segS;
  if (len == 0) return;
  const uint32_t gbase = batchOff[b] + segS;
  const uint32_t unit = unitArr[b];

  if (len <= CAP) {
    for (uint32_t j = lane; j < len; j += 32) seg[wave][j] = pairs[gbase + j];
    __builtin_amdgcn_wave_barrier();            // keep ds stores before ds loads
    asm volatile("" ::: "memory");              // (DS ops are in-order per wave)
    for (uint32_t e = lane; e < len; e += 32) {
      uint2 me = seg[wave][e];
      uint32_t c = 0;
      for (uint32_t j = 0; j < len; ++j) {
        uint2 o = seg[wave][j];
        c += (uint32_t)((o.y < me.y) | ((o.y == me.y) & (o.x < me.x)));
      }
      emit_event(ev, out, me.x, b, segS + c, unit);
    }
  } else {
    // pathological long bin: O(len^2) via L2-cached broadcast loads
    for (uint32_t e = lane; e < len; e += 32) {
      uint2 me = pairs[gbase + e];
      uint32_t c = 0;
      for (uint32_t j = 0; j < len; ++j) {
        uint2 o = pairs[gbase + j];
        c += (uint32_t)((o.y < me.y) | ((o.y == me.y) & (o.x < me.x)));
      }
      emit_event(ev, out, me.x, b, segS + c, unit);
    }
  }
}

// ---------- host-side launcher ----------
extern "C" void kernel_launch(void* const* d_in, const int* in_sizes, int n_in,
                              void* d_out, int out_size, void* d_ws, size_t ws_size,
                              hipStream_t stream) {
  const float* ev = (const float*)d_in[0];
  const uint32_t N = (uint32_t)(in_sizes[0] / 5);
  char* ws = (char*)d_ws;

  // workspace layout (~67.1 MB total)
  const size_t OFF_HIST     = 0;                        // B*NBIN u32 = 1 MB
  const size_t OFF_BINSTART = (size_t)1 << 20;          // 1 MB
  const size_t OFF_CURSOR   = (size_t)2 << 20;          // 1 MB
  const size_t OFF_SCALARS  = (size_t)3 << 20;          // counts/unit/batchOff
  const size_t OFF_PAIRS    = ((size_t)3 << 20) + 4096; // N * 8 B = 64 MB

  uint32_t* hist     = (uint32_t*)(ws + OFF_HIST);
  uint32_t* binStart = (uint32_t*)(ws + OFF_BINSTART);
  uint32_t* cursor   = (uint32_t*)(ws + OFF_CURSOR);
  uint32_t* counts   = (uint32_t*)(ws + OFF_SCALARS);
  uint32_t* unitArr  = counts + 16;
  uint32_t* batchOff = counts + 32;
  uint2*    pairs    = (uint2*)(ws + OFF_PAIRS);
  float*    out      = (float*)d_out;

  // zero tables + output (capture-safe memset nodes)
  hipMemsetAsync(ws, 0, OFF_PAIRS, stream);
  hipMemsetAsync(out, 0, (size_t)out_size * sizeof(float), stream);

  const uint32_t ntiles = (N + TILE - 1) / TILE;
  k_hist    <<<ntiles, HTHREADS, 0, stream>>>(ev, N, hist);
  k_scan    <<<B_, 1024, 0, stream>>>(hist, binStart, counts, unitArr);
  k_batchoff<<<1, 32, 0, stream>>>(counts, batchOff);
  k_scatter <<<ntiles, HTHREADS, 0, stream>>>(ev, N, binStart, batchOff, cursor, pairs);
  k_binsort <<<(B_ * NBIN) / 8, 256, 0, stream>>>(ev, pairs, binStart, counts, unitArr, batchOff, out);
}